// HadamardRotation_43533788512452
// MI455X (gfx1250) — compile-verified
//
#include <hip/hip_runtime.h>

typedef __attribute__((ext_vector_type(2))) float v2f;
typedef __attribute__((ext_vector_type(8))) float v8f;

#define WAVES_PER_BLOCK 2
#define APITCH 273   // per-'a' slab pitch (16*17 + 1) -> stage-1 store stride 273 (bank-free)
#define BPITCH 17    // per-'b' row pitch (16 + 1)     -> stride-17 accesses (bank-free)
#define LDS_WORDS_PER_WAVE (16 * APITCH)

// Walsh-Hadamard rotation of each length-4096 row, via H4096 = H16 (x) H16 (x) H16.
// One wave32 owns one row. 3 stages of 16 WMMA-tiles x 4 chained K=4 f32 WMMAs.
__global__ __launch_bounds__(WAVES_PER_BLOCK * 32)
void fwht4096_wmma_kernel(const float* __restrict__ x,
                          float* __restrict__ out,
                          int nrows)
{
    __shared__ float lds[WAVES_PER_BLOCK * LDS_WORDS_PER_WAVE]; // 34944 B

    const int lane = threadIdx.x & 31;
    const int wave = threadIdx.x >> 5;
    const int row  = blockIdx.x * WAVES_PER_BLOCK + wave;
    if (row >= nrows) return;          // wave-uniform: EXEC stays all-ones for WMMA

    float* __restrict__ w          = lds + wave * LDS_WORDS_PER_WAVE;
    const float* __restrict__ xr   = x   + (size_t)row * 4096;
    float* __restrict__ orow       = out + (size_t)row * 4096;

    const int li = lane & 15;   // N index (B/C/D lanes), M index (A lanes)
    const int lg = lane >> 4;   // lane-group: selects K pair (A/B), M+8 (C/D)

    // ---- B operand: unnormalized H16 (+-1) built in registers, no memory.
    // B(k,n): vgpr j holds k = 4q + 2*lg + j, n = li.  (H16 symmetric -> orientation-safe)
    v2f bm[4];
    #pragma unroll
    for (int q = 0; q < 4; ++q) {
        const int k0 = 4 * q + 2 * lg;
        bm[q].x = (__popc( k0      & li) & 1) ? -1.0f : 1.0f;
        bm[q].y = (__popc((k0 + 1) & li) & 1) ? -1.0f : 1.0f;
    }

    // ================= Stage 1: transform along axis a (stride 256) ===========
    // A(m=(b,c), k=a) = xr[k*256 + m], tile t covers m = 16t..16t+15 (b=t, c=mm).
    // Each global load instr: lanes 0-15 and 16-31 each read one full 64B line.
    #pragma unroll 4
    for (int t = 0; t < 16; ++t) {
        v8f d = {};
        #pragma unroll
        for (int q = 0; q < 4; ++q) {
            const int k0 = 4 * q + 2 * lg;
            v2f a;
            a.x = xr[(size_t)(k0    ) * 256 + t * 16 + li];
            a.y = xr[(size_t)(k0 + 1) * 256 + t * 16 + li];
            d = __builtin_amdgcn_wmma_f32_16x16x4_f32(false, a, false, bm[q],
                                                      (short)0, d, false, false);
        }
        // D(M=mm, N=a') -> w[a'][b=t][c=mm]; lanes 0-15 stride 273 -> 16 distinct banks
        #pragma unroll
        for (int v = 0; v < 8; ++v)
            w[li * APITCH + t * BPITCH + (v + 8 * lg)] = d[v];
    }
    asm volatile("" ::: "memory");   // same-wave LDS is in-order; just stop compiler reordering

    // ================= Stage 2: transform along axis b (stride 16) ============
    // Tile t = slab a=t.  A(mm=c, k=b) = w[t][k][mm]; in-place write w[t][b'][mm].
    #pragma unroll 4
    for (int t = 0; t < 16; ++t) {
        v8f d = {};
        #pragma unroll
        for (int q = 0; q < 4; ++q) {
            const int k0 = 4 * q + 2 * lg;
            v2f a;
            a.x = w[t * APITCH + (k0    ) * BPITCH + li];
            a.y = w[t * APITCH + (k0 + 1) * BPITCH + li];
            d = __builtin_amdgcn_wmma_f32_16x16x4_f32(false, a, false, bm[q],
                                                      (short)0, d, false, false);
        }
        #pragma unroll
        for (int v = 0; v < 8; ++v)
            w[t * APITCH + li * BPITCH + (v + 8 * lg)] = d[v];
    }
    asm volatile("" ::: "memory");

    // ================= Stage 3: transform along axis c (stride 1) =============
    // Tile t = slab a=t.  A(mm=b, k=c) = w[t][mm][k].  Output straight to global,
    // fold in total normalization (1/4)^3 = 1/64.
    #pragma unroll 4
    for (int t = 0; t < 16; ++t) {
        v8f d = {};
        #pragma unroll
        for (int q = 0; q < 4; ++q) {
            const int k0 = 4 * q + 2 * lg;
            v2f a;
            a.x = w[t * APITCH + li * BPITCH + k0    ];
            a.y = w[t * APITCH + li * BPITCH + k0 + 1];
            d = __builtin_amdgcn_wmma_f32_16x16x4_f32(false, a, false, bm[q],
                                                      (short)0, d, false, false);
        }
        // D(M=b, N=c') -> out[a=t, b, c'] ; each store instr writes two full 64B lines
        #pragma unroll
        for (int v = 0; v < 8; ++v)
            orow[t * 256 + (v + 8 * lg) * 16 + li] = d[v] * 0.015625f;
    }
}

extern "C" void kernel_launch(void* const* d_in, const int* in_sizes, int n_in,
                              void* d_out, int out_size, void* d_ws, size_t ws_size,
                              hipStream_t stream)
{
    const float* x = (const float*)d_in[0];
    // d_in[1] (H) is not needed: H is the fixed Sylvester Walsh-Hadamard matrix,
    // reconstructed exactly in-register from popcount parity.
    float* out = (float*)d_out;

    const int nrows = in_sizes[0] / 4096;                 // 4 * 4096 = 16384 rows
    dim3 block(WAVES_PER_BLOCK * 32);
    dim3 grid((nrows + WAVES_PER_BLOCK - 1) / WAVES_PER_BLOCK);
    hipLaunchKernelGGL(fwht4096_wmma_kernel, grid, block, 0, stream, x, out, nrows);
}